// ResampleToUVTexture_56410100465682
// MI455X (gfx1250) — compile-verified
//
#include <hip/hip_runtime.h>

// Problem constants (match reference)
#define HDIM   512
#define WDIM   1024
#define BDIM   8
#define CDIM   3
#define NFACES 80
#define GDIM   130
#define PPX    (GDIM * GDIM)        // 16900
#define NSAMP  (NFACES * PPX)       // 1,352,000 sample points
#define NPLANE (BDIM * CDIM)        // 24 image planes
#define PLANE  (HDIM * WDIM)        // 524,288 elements per plane

typedef float f2 __attribute__((ext_vector_type(2)));

// One thread per sample point: compute bilinear weights/offsets once,
// gather 4 texels from each of the 24 (b,c) planes, blend, NT-store.
__global__ __launch_bounds__(256) void bisph_unresample_kernel(
    const float* __restrict__ x,      // (B,C,H,W)
    const float* __restrict__ smap,   // (F,PP,2) flat
    float* __restrict__ out)          // (B,C,F,G,G) flat
{
    const int s = blockIdx.x * 256 + threadIdx.x;
    if (s >= NSAMP) return;

    // sample_map is touched exactly once per element -> non-temporal load
    // (th:TH_LOAD_NT keeps it from polluting caches that should hold x).
    const f2 uv = __builtin_nontemporal_load((const f2*)smap + s);
    const float xs = uv.x;
    const float ys = uv.y;

    const float x0f = floorf(xs);
    const float y0f = floorf(ys);
    const float wx  = xs - x0f;
    const float wy  = ys - y0f;

    const int x0 = (int)x0f;
    const int y0 = (int)y0f;

    // jnp.mod(x0, 1024) == (x0 & 1023) for two's-complement ints (W is pow2)
    const int x0w = x0 & (WDIM - 1);
    const int x1w = (x0 + 1) & (WDIM - 1);
    // jnp.clip(y0, 0, H-1)
    const int y0c = min(max(y0, 0), HDIM - 1);
    const int y1c = min(max(y0 + 1, 0), HDIM - 1);

    const float omwx = 1.0f - wx;
    const float omwy = 1.0f - wy;
    const float w00 = omwx * omwy;
    const float w01 = wx   * omwy;
    const float w10 = omwx * wy;
    const float w11 = wx   * wy;

    // Intra-plane texel offsets (shared by all 24 planes)
    const int o00 = y0c * WDIM + x0w;
    const int o01 = y0c * WDIM + x1w;
    const int o10 = y1c * WDIM + x0w;
    const int o11 = y1c * WDIM + x1w;

    // Gathers from x: regular-temporal loads -> x (48 MB) stays resident in
    // the 192 MB L2, where it is reused ~10x across faces.
    // Output is write-once, streamed -> non-temporal stores (th:TH_STORE_NT)
    // so the 124 MB output stream does not evict x.
#pragma unroll
    for (int p = 0; p < NPLANE; ++p) {
        const float* __restrict__ pl = x + (size_t)p * PLANE;
        float v = pl[o11] * w11;
        v = fmaf(pl[o10], w10, v);
        v = fmaf(pl[o01], w01, v);
        v = fmaf(pl[o00], w00, v);
        __builtin_nontemporal_store(v, out + (size_t)p * NSAMP + s);
    }
}

extern "C" void kernel_launch(void* const* d_in, const int* in_sizes, int n_in,
                              void* d_out, int out_size, void* d_ws, size_t ws_size,
                              hipStream_t stream)
{
    const float* x    = (const float*)d_in[0];   // 8*3*512*1024 floats
    const float* smap = (const float*)d_in[1];   // 80*16900*2 floats
    float* out        = (float*)d_out;           // 8*3*80*130*130 floats

    const int threads = 256;
    const int blocks  = (NSAMP + threads - 1) / threads; // 5282
    bisph_unresample_kernel<<<blocks, threads, 0, stream>>>(x, smap, out);
}